// PNANet_16527034155616
// MI455X (gfx1250) — compile-verified
//
#include <hip/hip_runtime.h>
#include <hip/hip_bf16.h>

typedef __attribute__((ext_vector_type(2))) float v2f;
typedef __attribute__((ext_vector_type(8))) float v8f;

// ---------------------------------------------------------------------------
// WMMA helper: D = A(16x4 f32) * B(4x16 f32) + C  (V_WMMA_F32_16X16X4_F32)
// ---------------------------------------------------------------------------
__device__ __forceinline__ v8f wmma_f32_k4(v2f a, v2f b, v8f c) {
    return __builtin_amdgcn_wmma_f32_16x16x4_f32(
        /*neg_a=*/false, a, /*neg_b=*/false, b,
        /*c_mod=*/(short)0, c, /*reuse_a=*/false, /*reuse_b=*/false);
}

// Native fire-and-forget float min/max atomics (lower to
// global_atomic_min_num_f32 / global_atomic_max_num_f32 on gfx1250,
// non-returning -> STOREcnt, RMW resolved at L2; no CAS retry loop).
__device__ __forceinline__ void atomicMinFloat(float* addr, float val) {
    (void)atomicMin(addr, val);
}
__device__ __forceinline__ void atomicMaxFloat(float* addr, float val) {
    (void)atomicMax(addr, val);
}

// ---------------------------------------------------------------------------
// Utility kernels
// ---------------------------------------------------------------------------
__global__ void fill_kernel(float* __restrict__ p, float v, long long n) {
    long long i = (long long)blockIdx.x * blockDim.x + threadIdx.x;
    if (i < n) p[i] = v;
}

__global__ void deg_kernel(const int* __restrict__ col, float* __restrict__ deg, int E_) {
    int e = blockIdx.x * blockDim.x + threadIdx.x;
    if (e < E_) atomicAdd(&deg[col[e]], 1.0f);
}

__global__ void logsum_kernel(const float* __restrict__ deg, float* __restrict__ avg, int N_) {
    int i = blockIdx.x * blockDim.x + threadIdx.x;
    float v = (i < N_) ? logf(deg[i] + 1.0f) : 0.0f;
    __shared__ float sh[256];
    sh[threadIdx.x] = v;
    __syncthreads();
    for (int off = 128; off > 0; off >>= 1) {
        if (threadIdx.x < off) sh[threadIdx.x] += sh[threadIdx.x + off];
        __syncthreads();
    }
    if (threadIdx.x == 0) atomicAdd(&avg[0], sh[0]);
}

__global__ void avg_final_kernel(float* __restrict__ avg, int N_) {
    avg[1] = avg[0] / (float)N_;
}

// ---------------------------------------------------------------------------
// e_enc = edge_attr(E x 32) @ We(32 x 64) + be    — one wave per 16 edges
// ---------------------------------------------------------------------------
__global__ void eenc_kernel(const float* __restrict__ ea, const float* __restrict__ We,
                            const float* __restrict__ be, float* __restrict__ eenc, int E_) {
    int wave = (int)((blockIdx.x * (long long)blockDim.x + threadIdx.x) >> 5);
    int lane = threadIdx.x & 31;
    int base = wave * 16;
    if (base >= E_) return;                 // wave-uniform
    int m = lane & 15;
    int hl = lane >> 4;
    const float* arow = ea + (size_t)(base + m) * 32;
    v8f acc[4] = {};
    for (int kk = 0; kk < 8; ++kk) {
        int k0 = kk * 4 + 2 * hl;
        v2f A; A.x = arow[k0]; A.y = arow[k0 + 1];
#pragma unroll
        for (int c = 0; c < 4; ++c) {
            v2f B;
            B.x = We[(size_t)k0 * 64 + c * 16 + m];
            B.y = We[(size_t)(k0 + 1) * 64 + c * 16 + m];
            acc[c] = wmma_f32_k4(A, B, acc[c]);
        }
    }
#pragma unroll
    for (int c = 0; c < 4; ++c) {
        int f = c * 16 + m;
        float bias = be[f];
#pragma unroll
        for (int v = 0; v < 8; ++v) {
            int r = v + 8 * hl;
            eenc[(size_t)(base + r) * 64 + f] = acc[c][v] + bias;
        }
    }
}

// ---------------------------------------------------------------------------
// Edge messages: h = [x_i | x_j | e_enc](192) @ preW_t(192x64) + preb_t,
// fused with atomic segment {sum, sumsq, min, max} into per-node stats.
// One wave per 16 edges.
// ---------------------------------------------------------------------------
template <int T>
__global__ void edge_msg_kernel(const float* __restrict__ x,
                                const int* __restrict__ row, const int* __restrict__ col,
                                const float* __restrict__ eenc,
                                const float* __restrict__ preW, const float* __restrict__ preb,
                                float* __restrict__ s_sum, float* __restrict__ s_sq,
                                float* __restrict__ s_mn, float* __restrict__ s_mx, int E_) {
    int wave = (int)((blockIdx.x * (long long)blockDim.x + threadIdx.x) >> 5);
    int lane = threadIdx.x & 31;
    int base = wave * 16;
    if (base >= E_) return;                 // wave-uniform
    int m = lane & 15;
    int hl = lane >> 4;
    int e_lane = base + m;
    const float* ai = x + (size_t)col[e_lane] * 64;   // x_i = x[col]
    const float* aj = x + (size_t)row[e_lane] * 64;   // x_j = x[row]
    const float* ae = eenc + (size_t)e_lane * 64;
    for (int t = 0; t < T; ++t) {
        const float* W = preW + (size_t)t * 192 * 64;
        v8f acc[4] = {};
        for (int kk = 0; kk < 48; ++kk) {
            int ksec = kk >> 4;
            int klocal = (kk & 15) * 4 + 2 * hl;
            const float* arow = (ksec == 0) ? ai : (ksec == 1) ? aj : ae;
            v2f A; A.x = arow[klocal]; A.y = arow[klocal + 1];
            int wrow = kk * 4 + 2 * hl;
#pragma unroll
            for (int c = 0; c < 4; ++c) {
                v2f B;
                B.x = W[(size_t)wrow * 64 + c * 16 + m];
                B.y = W[(size_t)(wrow + 1) * 64 + c * 16 + m];
                acc[c] = wmma_f32_k4(A, B, acc[c]);
            }
        }
#pragma unroll
        for (int c = 0; c < 4; ++c) {
            int f = c * 16 + m;
            float bias = preb[t * 64 + f];
#pragma unroll
            for (int v = 0; v < 8; ++v) {
                int e = base + v + 8 * hl;
                int cn = col[e];
                float val = acc[c][v] + bias;
                size_t idx = ((size_t)cn * T + t) * 64 + f;
                atomicAdd(&s_sum[idx], val);
                atomicAdd(&s_sq[idx], val * val);
                atomicMinFloat(&s_mn[idx], val);
                atomicMaxFloat(&s_mx[idx], val);
            }
        }
    }
}

// ---------------------------------------------------------------------------
// Node post: build cat(832) per node in LDS, GEMM with postW_t(832xG)+postb,
// concat towers, then lin(64x64)+linb.  Block = 128 threads, 16 nodes.
// ---------------------------------------------------------------------------
template <int T, int G>
__global__ void node_post_kernel(const float* __restrict__ xin,
                                 const float* __restrict__ deg, const float* __restrict__ avgp,
                                 const float* __restrict__ s_sum, const float* __restrict__ s_sq,
                                 const float* __restrict__ s_mn, const float* __restrict__ s_mx,
                                 const float* __restrict__ postW, const float* __restrict__ postb,
                                 const float* __restrict__ linW, const float* __restrict__ linb,
                                 float* __restrict__ outp, int N_) {
    __shared__ float cat[16 * 832];
    __shared__ float flat[16 * 64];
    int base = blockIdx.x * 16;
    int tid = threadIdx.x;
    int lane = tid & 31, wid = tid >> 5;
    float avg_log = avgp[1];
    for (int t = 0; t < T; ++t) {
        for (int idx = tid; idx < 16 * 64; idx += 128) {
            int r = idx >> 6, f = idx & 63;
            int n = base + r;
            float dg = deg[n];
            float d = fmaxf(dg, 1.0f);
            size_t sidx = ((size_t)n * T + t) * 64 + f;
            float mean  = s_sum[sidx] / d;
            float mean2 = s_sq[sidx] / d;
            float stdv  = sqrtf(fmaxf(mean2 - mean * mean, 0.0f) + 1e-5f);
            bool has = dg > 0.0f;
            float mn = has ? s_mn[sidx] : 0.0f;
            float mx = has ? s_mx[sidx] : 0.0f;
            float lg = logf(d + 1.0f);
            float amp = lg / avg_log;
            float att = avg_log / lg;
            float* cr = &cat[r * 832];
            cr[f]        = xin[(size_t)n * 64 + f];
            cr[64 + f]   = mean;       cr[128 + f] = mn;       cr[192 + f] = mx;       cr[256 + f] = stdv;
            cr[320 + f]  = mean * amp; cr[384 + f] = mn * amp; cr[448 + f] = mx * amp; cr[512 + f] = stdv * amp;
            cr[576 + f]  = mean * att; cr[640 + f] = mn * att; cr[704 + f] = mx * att; cr[768 + f] = stdv * att;
        }
        __syncthreads();
        constexpr int CT = G / 16;
        if (wid < CT) {                     // wave-uniform
            int c = wid;
            int m = lane & 15, hl = lane >> 4;
            const float* W = postW + (size_t)t * 832 * G;
            const float* arow = &cat[m * 832];
            v8f acc = {};
            for (int kk = 0; kk < 208; ++kk) {
                int k0 = kk * 4 + 2 * hl;
                v2f A; A.x = arow[k0]; A.y = arow[k0 + 1];
                v2f B;
                B.x = W[(size_t)k0 * G + c * 16 + m];
                B.y = W[(size_t)(k0 + 1) * G + c * 16 + m];
                acc = wmma_f32_k4(A, B, acc);
            }
            int f = c * 16 + m;
            float bias = postb[t * G + f];
#pragma unroll
            for (int v = 0; v < 8; ++v) {
                int r = v + 8 * hl;
                flat[r * 64 + t * G + f] = acc[v] + bias;
            }
        }
        __syncthreads();
    }
    // lin: (16 x 64) @ linW(64 x 64) + linb
    {
        int c = wid;
        int m = lane & 15, hl = lane >> 4;
        const float* arow = &flat[m * 64];
        v8f acc = {};
        for (int kk = 0; kk < 16; ++kk) {
            int k0 = kk * 4 + 2 * hl;
            v2f A; A.x = arow[k0]; A.y = arow[k0 + 1];
            v2f B;
            B.x = linW[k0 * 64 + c * 16 + m];
            B.y = linW[(k0 + 1) * 64 + c * 16 + m];
            acc = wmma_f32_k4(A, B, acc);
        }
        int f = c * 16 + m;
        float bias = linb[f];
#pragma unroll
        for (int v = 0; v < 8; ++v) {
            int n = base + v + 8 * hl;
            if (n < N_) outp[(size_t)n * 64 + f] = acc[v] + bias;
        }
    }
}

// ---------------------------------------------------------------------------
// Batchnorm
// ---------------------------------------------------------------------------
__global__ void bn_stats_kernel(const float* __restrict__ xin, float* __restrict__ meanp,
                                float* __restrict__ rstdp, int N_) {
    int f = blockIdx.x;
    int tid = threadIdx.x;
    float s = 0.f, sq = 0.f;
    for (int n = tid; n < N_; n += 256) {
        float v = xin[(size_t)n * 64 + f];
        s += v; sq += v * v;
    }
    __shared__ float sh[256], shq[256];
    sh[tid] = s; shq[tid] = sq;
    __syncthreads();
    for (int off = 128; off > 0; off >>= 1) {
        if (tid < off) { sh[tid] += sh[tid + off]; shq[tid] += shq[tid + off]; }
        __syncthreads();
    }
    if (tid == 0) {
        float m = sh[0] / (float)N_;
        float var = shq[0] / (float)N_ - m * m;
        meanp[f] = m;
        rstdp[f] = rsqrtf(fmaxf(var, 0.0f) + 1e-5f);
    }
}

__global__ void bn_apply_kernel(const float* __restrict__ xin, const float* __restrict__ meanp,
                                const float* __restrict__ rstdp, const float* __restrict__ g,
                                const float* __restrict__ b, float* __restrict__ outp, int total) {
    int i = blockIdx.x * blockDim.x + threadIdx.x;
    if (i >= total) return;
    int f = i & 63;
    float v = (xin[i] - meanp[f]) * rstdp[f] * g[f] + b[f];
    outp[i] = fmaxf(v, 0.0f);
}

// ---------------------------------------------------------------------------
// Pooling: nsum[col] += h[row]; score = nsum@Wrel + brel + h@Wroot
// ---------------------------------------------------------------------------
__global__ void nsum_kernel(const float* __restrict__ h, const int* __restrict__ row,
                            const int* __restrict__ col, float* __restrict__ nsum, int E_) {
    long long i = (long long)blockIdx.x * blockDim.x + threadIdx.x;
    if (i >= (long long)E_ * 64) return;
    int e = (int)(i >> 6), f = (int)(i & 63);
    atomicAdd(&nsum[(size_t)col[e] * 64 + f], h[(size_t)row[e] * 64 + f]);
}

__global__ void score_kernel(const float* __restrict__ h, const float* __restrict__ nsum,
                             const float* __restrict__ Wrel, const float* __restrict__ brel,
                             const float* __restrict__ Wroot, float* __restrict__ score, int N_) {
    int n = blockIdx.x * blockDim.x + threadIdx.x;
    if (n >= N_) return;
    float s = 0.f;
    for (int f = 0; f < 64; ++f)
        s += nsum[(size_t)n * 64 + f] * Wrel[f] + h[(size_t)n * 64 + f] * Wroot[f];
    score[n] = s + brel[0];
}

// ---------------------------------------------------------------------------
// Top-K via 5 rounds of two-stage argmax (lowest index wins ties, like top_k)
// ---------------------------------------------------------------------------
__global__ void argmax_part_kernel(const float* __restrict__ score, float* __restrict__ pv,
                                   int* __restrict__ pi, int N_) {
    int tid = threadIdx.x;
    float best = -3.4e38f; int bi = 0x7fffffff;
    for (int n = blockIdx.x * 256 + tid; n < N_; n += gridDim.x * 256) {
        float v = score[n];
        if (v > best || (v == best && n < bi)) { best = v; bi = n; }
    }
    __shared__ float sv[256]; __shared__ int si[256];
    sv[tid] = best; si[tid] = bi;
    __syncthreads();
    for (int off = 128; off > 0; off >>= 1) {
        if (tid < off) {
            if (sv[tid + off] > sv[tid] || (sv[tid + off] == sv[tid] && si[tid + off] < si[tid])) {
                sv[tid] = sv[tid + off]; si[tid] = si[tid + off];
            }
        }
        __syncthreads();
    }
    if (tid == 0) { pv[blockIdx.x] = sv[0]; pi[blockIdx.x] = si[0]; }
}

__global__ void argmax_final_kernel(float* __restrict__ score, const float* __restrict__ pv,
                                    const int* __restrict__ pi, int parts,
                                    float* __restrict__ topv, int* __restrict__ perm, int k) {
    int tid = threadIdx.x;
    __shared__ float sv[256]; __shared__ int si[256];
    sv[tid] = (tid < parts) ? pv[tid] : -3.4e38f;
    si[tid] = (tid < parts) ? pi[tid] : 0x7fffffff;
    __syncthreads();
    for (int off = 128; off > 0; off >>= 1) {
        if (tid < off) {
            if (sv[tid + off] > sv[tid] || (sv[tid + off] == sv[tid] && si[tid + off] < si[tid])) {
                sv[tid] = sv[tid + off]; si[tid] = si[tid + off];
            }
        }
        __syncthreads();
    }
    if (tid == 0) {
        topv[k] = sv[0];
        perm[k] = si[0];
        score[si[0]] = -3.4e38f;   // exclude from next round
    }
}

// ---------------------------------------------------------------------------
// Final tiny MLP on the 5 selected nodes
// ---------------------------------------------------------------------------
__global__ void mlp_kernel(const float* __restrict__ h, const int* __restrict__ perm,
                           const float* __restrict__ topv,
                           const float* __restrict__ mW1, const float* __restrict__ mb1,
                           const float* __restrict__ mW2, const float* __restrict__ mb2,
                           const float* __restrict__ mW3, const float* __restrict__ mb3,
                           float* __restrict__ out) {
    __shared__ float xp[5][64], z1[5][32], z2[5][16];
    int tid = threadIdx.x;
    for (int i = tid; i < 5 * 64; i += blockDim.x) {
        int r = i >> 6, f = i & 63;
        xp[r][f] = h[(size_t)perm[r] * 64 + f] * tanhf(topv[r]);
    }
    __syncthreads();
    for (int i = tid; i < 5 * 32; i += blockDim.x) {
        int r = i / 32, j = i % 32;
        float s = mb1[j];
        for (int f = 0; f < 64; ++f) s += xp[r][f] * mW1[f * 32 + j];
        z1[r][j] = fmaxf(s, 0.0f);
    }
    __syncthreads();
    for (int i = tid; i < 5 * 16; i += blockDim.x) {
        int r = i / 16, j = i % 16;
        float s = mb2[j];
        for (int f = 0; f < 32; ++f) s += z1[r][f] * mW2[f * 16 + j];
        z2[r][j] = fmaxf(s, 0.0f);
    }
    __syncthreads();
    for (int i = tid; i < 5 * 2; i += blockDim.x) {
        int r = i / 2, j = i % 2;
        float s = mb3[j];
        for (int f = 0; f < 16; ++f) s += z2[r][f] * mW3[f * 2 + j];
        out[r * 2 + j] = s;
    }
}

// ---------------------------------------------------------------------------
// Host driver
// ---------------------------------------------------------------------------
extern "C" void kernel_launch(void* const* d_in, const int* in_sizes, int n_in,
                              void* d_out, int out_size, void* d_ws, size_t ws_size,
                              hipStream_t stream) {
    constexpr int NN = 50000, EE = 400000, TMAX = 4, KTOP = 5;

    const float* x        = (const float*)d_in[0];
    const int*   eidx     = (const int*)d_in[1];
    const float* eattr    = (const float*)d_in[2];
    const float* We1      = (const float*)d_in[4];
    const float* be1      = (const float*)d_in[5];
    const float* preW1    = (const float*)d_in[6];
    const float* preb1    = (const float*)d_in[7];
    const float* postW1   = (const float*)d_in[8];
    const float* postb1   = (const float*)d_in[9];
    const float* linW1    = (const float*)d_in[10];
    const float* linb1    = (const float*)d_in[11];
    const float* bn1g     = (const float*)d_in[12];
    const float* bn1b     = (const float*)d_in[13];
    const float* We2      = (const float*)d_in[14];
    const float* be2      = (const float*)d_in[15];
    const float* preW2    = (const float*)d_in[16];
    const float* preb2    = (const float*)d_in[17];
    const float* postW2   = (const float*)d_in[18];
    const float* postb2   = (const float*)d_in[19];
    const float* linW2    = (const float*)d_in[20];
    const float* linb2    = (const float*)d_in[21];
    const float* bn2g     = (const float*)d_in[22];
    const float* bn2b     = (const float*)d_in[23];
    const float* poolWrel = (const float*)d_in[24];
    const float* poolbrel = (const float*)d_in[25];
    const float* poolWroot= (const float*)d_in[26];
    const float* mW1      = (const float*)d_in[27];
    const float* mb1      = (const float*)d_in[28];
    const float* mW2      = (const float*)d_in[29];
    const float* mb2      = (const float*)d_in[30];
    const float* mW3      = (const float*)d_in[31];
    const float* mb3      = (const float*)d_in[32];

    const int* row = eidx;
    const int* col = eidx + EE;

    float* ws = (float*)d_ws;
    size_t o = 0;
    float* w_deg   = ws + o; o += NN;
    float* w_avg   = ws + o; o += 2;
    float* w_mean  = ws + o; o += 64;
    float* w_rstd  = ws + o; o += 64;
    float* w_topv  = ws + o; o += 8;
    int*   w_perm  = (int*)(ws + o); o += 8;
    float* w_pv    = ws + o; o += 4096;
    int*   w_pi    = (int*)(ws + o); o += 4096;
    float* w_score = ws + o; o += NN;
    float* w_prebn = ws + o; o += (size_t)NN * 64;
    float* w_h     = ws + o; o += (size_t)NN * 64;
    float* w_nsum  = ws + o; o += (size_t)NN * 64;
    float* w_eenc  = ws + o; o += (size_t)EE * 64;
    float* w_sum   = ws + o; o += (size_t)NN * TMAX * 64;
    float* w_sq    = ws + o; o += (size_t)NN * TMAX * 64;
    float* w_mn    = ws + o; o += (size_t)NN * TMAX * 64;
    float* w_mx    = ws + o; o += (size_t)NN * TMAX * 64;

    auto fill = [&](float* p, float v, long long n) {
        int blocks = (int)((n + 255) / 256);
        fill_kernel<<<blocks, 256, 0, stream>>>(p, v, n);
    };

    const int edgeWaveBlocks = (EE / 16 * 32 + 255) / 256;  // 3125
    const int nodeBlocks = NN / 16;                          // 3125

    // degree + avg_log
    fill(w_deg, 0.0f, NN);
    deg_kernel<<<(EE + 255) / 256, 256, 0, stream>>>(col, w_deg, EE);
    fill(w_avg, 0.0f, 2);
    logsum_kernel<<<(NN + 255) / 256, 256, 0, stream>>>(w_deg, w_avg, NN);
    avg_final_kernel<<<1, 1, 0, stream>>>(w_avg, NN);

    // ------------------- Layer 1 (T=1, G=64) -------------------
    {
        long long statn = (long long)NN * 1 * 64;
        fill(w_sum, 0.0f, statn); fill(w_sq, 0.0f, statn);
        fill(w_mn, 3.0e38f, statn); fill(w_mx, -3.0e38f, statn);
        eenc_kernel<<<edgeWaveBlocks, 256, 0, stream>>>(eattr, We1, be1, w_eenc, EE);
        edge_msg_kernel<1><<<edgeWaveBlocks, 256, 0, stream>>>(
            x, row, col, w_eenc, preW1, preb1, w_sum, w_sq, w_mn, w_mx, EE);
        node_post_kernel<1, 64><<<nodeBlocks, 128, 0, stream>>>(
            x, w_deg, w_avg, w_sum, w_sq, w_mn, w_mx,
            postW1, postb1, linW1, linb1, w_prebn, NN);
        bn_stats_kernel<<<64, 256, 0, stream>>>(w_prebn, w_mean, w_rstd, NN);
        bn_apply_kernel<<<(NN * 64 + 255) / 256, 256, 0, stream>>>(
            w_prebn, w_mean, w_rstd, bn1g, bn1b, w_h, NN * 64);
    }

    // ------------------- Layer 2 (T=4, G=16) -------------------
    {
        long long statn = (long long)NN * 4 * 64;
        fill(w_sum, 0.0f, statn); fill(w_sq, 0.0f, statn);
        fill(w_mn, 3.0e38f, statn); fill(w_mx, -3.0e38f, statn);
        eenc_kernel<<<edgeWaveBlocks, 256, 0, stream>>>(eattr, We2, be2, w_eenc, EE);
        edge_msg_kernel<4><<<edgeWaveBlocks, 256, 0, stream>>>(
            w_h, row, col, w_eenc, preW2, preb2, w_sum, w_sq, w_mn, w_mx, EE);
        node_post_kernel<4, 16><<<nodeBlocks, 128, 0, stream>>>(
            w_h, w_deg, w_avg, w_sum, w_sq, w_mn, w_mx,
            postW2, postb2, linW2, linb2, w_prebn, NN);
        bn_stats_kernel<<<64, 256, 0, stream>>>(w_prebn, w_mean, w_rstd, NN);
        bn_apply_kernel<<<(NN * 64 + 255) / 256, 256, 0, stream>>>(
            w_prebn, w_mean, w_rstd, bn2g, bn2b, w_h, NN * 64);
    }

    // ------------------- Pooling + top-K + MLP -------------------
    fill(w_nsum, 0.0f, (long long)NN * 64);
    nsum_kernel<<<(int)(((long long)EE * 64 + 255) / 256), 256, 0, stream>>>(
        w_h, row, col, w_nsum, EE);
    score_kernel<<<(NN + 255) / 256, 256, 0, stream>>>(
        w_h, w_nsum, poolWrel, poolbrel, poolWroot, w_score, NN);
    for (int k = 0; k < KTOP; ++k) {
        argmax_part_kernel<<<256, 256, 0, stream>>>(w_score, w_pv, w_pi, NN);
        argmax_final_kernel<<<1, 256, 0, stream>>>(w_score, w_pv, w_pi, 256, w_topv, w_perm, k);
    }
    mlp_kernel<<<1, 256, 0, stream>>>(w_h, w_perm, w_topv,
                                      mW1, mb1, mW2, mb2, mW3, mb3, (float*)d_out);
}